// RGAT_15221364097427
// MI455X (gfx1250) — compile-verified
//
#include <hip/hip_runtime.h>
#include <hip/hip_bf16.h>

// ---------------------------------------------------------------------------
// RGAT (2-layer, basis-decomposed, additive attention) for MI455X / gfx1250.
// N=50000 nodes, D=100, B=4 bases, E=400000 edges, R=474 relations.
//
// CSR (built once, shared by both layers, dst-sorted):
//   zero -> hist -> scan (single-WG LDS scan) -> fill
// Per layer:
//   1. qb/kb  = fold q,k into bases                       (tiny)
//   2. xb     = x @ W  via V_WMMA_F32_16X16X4_F32         (5 tiles / wave)
//   3. xq/xk  = x @ qb^T, x @ kb^T
//   4. aggregate: wave32 per dst node -- softmax (shfl reductions) +
//      register-resident weighted accumulation + bias (+relu). NO float atomics.
// xb (80MB) is L2-resident (192MB L2) so all edge gathers hit L2; removing the
// 80M global_atomic_add_f32 of the naive scatter removes the atomic-rate wall.
// ---------------------------------------------------------------------------

typedef float v2f __attribute__((ext_vector_type(2)));
typedef float v8f __attribute__((ext_vector_type(8)));

#define DIMD 100
#define BD   400        // B * D

// ---- 1. fold q,k into bases: qb[b*100+i] = sum_o basis[b,i,o]*q[o] --------
__global__ void rgat_qbkb(const float* __restrict__ basis,
                          const float* __restrict__ q,
                          const float* __restrict__ k,
                          float* __restrict__ qb, float* __restrict__ kb) {
    int t = blockIdx.x * blockDim.x + threadIdx.x;       // t = b*100 + i
    if (t >= BD) return;
    const float* brow = basis + (size_t)t * DIMD;
    float sq = 0.f, sk = 0.f;
    #pragma unroll 4
    for (int o = 0; o < DIMD; ++o) {
        float w = brow[o];
        sq += w * q[o];
        sk += w * k[o];
    }
    qb[t] = sq;
    kb[t] = sk;
}

// ---- 2. xb[n, b*100+o] = sum_i x[n,i]*basis[b,i,o]  (WMMA f32 16x16x4) ----
// One wave computes a 16x80 slab: 5 accumulators share one A fragment.
// rowTiles = N/16 (=3125), colGroups = 5 (25 tiles of 16 over 400 cols).
// All dims divide evenly -> EXEC stays all-1s through every WMMA.
__global__ void rgat_gemm_xb(const float* __restrict__ x,
                             const int*   __restrict__ gather,   // null for layer 2
                             const float* __restrict__ basis,
                             float* __restrict__ xb,
                             int nRowTiles) {
    const int wavesPerBlock = blockDim.x >> 5;
    int wave = blockIdx.x * wavesPerBlock + (threadIdx.x >> 5);
    const int lane = threadIdx.x & 31;
    const int totalWaves = nRowTiles * 5;
    if (wave >= totalWaves) return;                      // whole-wave uniform

    const int rowTile = wave / 5;
    const int cg      = wave % 5;                        // col group: cols [cg*80, cg*80+80)
    const int r0 = rowTile * 16;

    // A fragment (ISA 16x4 f32 layout):
    //   lanes 0-15 : M=lane,    VGPR0=K(k0+0), VGPR1=K(k0+1)
    //   lanes 16-31: M=lane-16, VGPR0=K(k0+2), VGPR1=K(k0+3)
    const int mA   = lane & 15;
    const int kOff = (lane >> 4) << 1;                   // 0 or 2
    int rowA = r0 + mA;
    if (gather) rowA = gather[rowA];
    const float* xrow = x + (size_t)rowA * DIMD;

    // B fragment column pointers: W[k][c] = basis[b*10000 + k*100 + o], c=b*100+o
    const float* w0; const float* w1; const float* w2; const float* w3; const float* w4;
    {
        int cbase = cg * 80 + (lane & 15);
        int c;
        c = cbase;      w0 = basis + (size_t)(c / DIMD) * (DIMD * DIMD) + (c % DIMD);
        c = cbase + 16; w1 = basis + (size_t)(c / DIMD) * (DIMD * DIMD) + (c % DIMD);
        c = cbase + 32; w2 = basis + (size_t)(c / DIMD) * (DIMD * DIMD) + (c % DIMD);
        c = cbase + 48; w3 = basis + (size_t)(c / DIMD) * (DIMD * DIMD) + (c % DIMD);
        c = cbase + 64; w4 = basis + (size_t)(c / DIMD) * (DIMD * DIMD) + (c % DIMD);
    }

    v8f acc0 = {}, acc1 = {}, acc2 = {}, acc3 = {}, acc4 = {};
    #pragma unroll
    for (int k0 = 0; k0 < DIMD; k0 += 4) {
        const int ka = k0 + kOff;
        v2f a;
        a.x = xrow[ka];
        a.y = xrow[ka + 1];
        v2f b0, b1, b2, b3, b4;
        b0.x = w0[(size_t)ka * DIMD]; b0.y = w0[(size_t)(ka + 1) * DIMD];
        b1.x = w1[(size_t)ka * DIMD]; b1.y = w1[(size_t)(ka + 1) * DIMD];
        b2.x = w2[(size_t)ka * DIMD]; b2.y = w2[(size_t)(ka + 1) * DIMD];
        b3.x = w3[(size_t)ka * DIMD]; b3.y = w3[(size_t)(ka + 1) * DIMD];
        b4.x = w4[(size_t)ka * DIMD]; b4.y = w4[(size_t)(ka + 1) * DIMD];
        acc0 = __builtin_amdgcn_wmma_f32_16x16x4_f32(false, a, false, b0, (short)0, acc0, false, false);
        acc1 = __builtin_amdgcn_wmma_f32_16x16x4_f32(false, a, false, b1, (short)0, acc1, false, false);
        acc2 = __builtin_amdgcn_wmma_f32_16x16x4_f32(false, a, false, b2, (short)0, acc2, false, false);
        acc3 = __builtin_amdgcn_wmma_f32_16x16x4_f32(false, a, false, b3, (short)0, acc3, false, false);
        acc4 = __builtin_amdgcn_wmma_f32_16x16x4_f32(false, a, false, b4, (short)0, acc4, false, false);
    }

    // D layout: VGPR r -> lanes 0-15: (M=r, N=lane); lanes 16-31: (M=r+8, N=lane-16)
    const int mOut = (lane >> 4) * 8;
    const int nOut = lane & 15;
    float* obase = xb + (size_t)r0 * BD + cg * 80 + nOut;
    #pragma unroll
    for (int r = 0; r < 8; ++r) {
        float* row = obase + (size_t)(mOut + r) * BD;
        row[0]  = acc0[r];
        row[16] = acc1[r];
        row[32] = acc2[r];
        row[48] = acc3[r];
        row[64] = acc4[r];
    }
}

// ---- 3. xq[n,b] = x[n,:]·qb[b,:]  (and xk) --------------------------------
__global__ void rgat_xqxk(const float* __restrict__ x,
                          const int*   __restrict__ gather,
                          const float* __restrict__ qb,
                          const float* __restrict__ kb,
                          float* __restrict__ xq, float* __restrict__ xk,
                          int n) {
    int t = blockIdx.x * blockDim.x + threadIdx.x;
    if (t >= n) return;
    int row = gather ? gather[t] : t;
    const float* xr = x + (size_t)row * DIMD;
    float aq0 = 0, aq1 = 0, aq2 = 0, aq3 = 0;
    float ak0 = 0, ak1 = 0, ak2 = 0, ak3 = 0;
    #pragma unroll 4
    for (int i = 0; i < DIMD; ++i) {
        float xv = xr[i];
        aq0 += xv * qb[i];           ak0 += xv * kb[i];
        aq1 += xv * qb[DIMD + i];    ak1 += xv * kb[DIMD + i];
        aq2 += xv * qb[2*DIMD + i];  ak2 += xv * kb[2*DIMD + i];
        aq3 += xv * qb[3*DIMD + i];  ak3 += xv * kb[3*DIMD + i];
    }
    xq[t*4+0] = aq0; xq[t*4+1] = aq1; xq[t*4+2] = aq2; xq[t*4+3] = aq3;
    xk[t*4+0] = ak0; xk[t*4+1] = ak1; xk[t*4+2] = ak2; xk[t*4+3] = ak3;
}

// ---- CSR build (once; both layers share the graph) ------------------------
__global__ void rgat_csr_zero(int* __restrict__ cnt, int n) {
    int t = blockIdx.x * blockDim.x + threadIdx.x;
    if (t < n) cnt[t] = 0;
}

__global__ void rgat_csr_hist(const int* __restrict__ dst, int* __restrict__ cnt, int nE) {
    int e = blockIdx.x * blockDim.x + threadIdx.x;
    if (e < nE) atomicAdd(&cnt[dst[e]], 1);
}

// single-workgroup exclusive scan over n counts -> rowptr[0..n], cursor copy
__global__ void rgat_csr_scan(const int* __restrict__ cnt,
                              int* __restrict__ rowptr, int* __restrict__ cursor,
                              int n) {
    __shared__ int smem[1024];
    __shared__ int carry;
    const int tid = threadIdx.x;
    if (tid == 0) carry = 0;
    for (int base = 0; base < n; base += 1024) {
        __syncthreads();                       // carry ready / smem free
        int myCarry = carry;
        int i = base + tid;
        int v = (i < n) ? cnt[i] : 0;
        smem[tid] = v;
        __syncthreads();
        #pragma unroll
        for (int off = 1; off < 1024; off <<= 1) {
            int t = (tid >= off) ? smem[tid - off] : 0;
            __syncthreads();
            smem[tid] += t;
            __syncthreads();
        }
        int excl = myCarry + smem[tid] - v;
        if (i < n) { rowptr[i] = excl; cursor[i] = excl; }
        __syncthreads();
        if (tid == 0) carry = myCarry + smem[1023];
    }
    __syncthreads();
    if (tid == 0) rowptr[n] = carry;
}

__global__ void rgat_csr_fill(const int* __restrict__ dst, int* __restrict__ cursor,
                              int* __restrict__ eord, int nE) {
    int e = blockIdx.x * blockDim.x + threadIdx.x;
    if (e < nE) {
        int p = atomicAdd(&cursor[dst[e]], 1);
        eord[p] = e;
    }
}

// ---- 4. aggregate: wave32 per destination node ----------------------------
// Phase A: lanes over edges -> logits + wave-max. Phase B: exp + wave-sum.
// Phase C: edges serial, lanes over D=100; output row kept in registers;
// bias (+relu) folded in; single plain store -- zero float atomics.
__global__ void rgat_aggregate(const int* __restrict__ rowptr,
                               const int* __restrict__ eord,
                               const int* __restrict__ src,
                               const int* __restrict__ et,
                               const float* __restrict__ att,   // [R,4]
                               const float* __restrict__ xq,    // [N,4]
                               const float* __restrict__ xk,    // [N,4]
                               const float* __restrict__ xb,    // [N,400] (L2)
                               const float* __restrict__ bias,  // [100]
                               float* __restrict__ tl,          // [E] scratch
                               float* __restrict__ out,         // [N,100]
                               int n, int doRelu) {
    int d    = blockIdx.x * (blockDim.x >> 5) + (threadIdx.x >> 5);
    int lane = threadIdx.x & 31;
    if (d >= n) return;                                  // whole-wave uniform

    const int start = rowptr[d];
    const int end   = rowptr[d + 1];
    const float4 qd = ((const float4*)xq)[d];

    // A: logits + segment max
    float m = -INFINITY;
    for (int idx = start + lane; idx < end; idx += 32) {
        int e = eord[idx];
        float4 a  = ((const float4*)att)[et[e]];
        float4 ks = ((const float4*)xk)[src[e]];
        float v = a.x * (qd.x + ks.x) + a.y * (qd.y + ks.y)
                + a.z * (qd.z + ks.z) + a.w * (qd.w + ks.w);
        v = (v > 0.f) ? v : 0.2f * v;                    // leaky_relu, slope 0.2
        tl[idx] = v;
        m = fmaxf(m, v);
    }
    #pragma unroll
    for (int off = 16; off; off >>= 1) m = fmaxf(m, __shfl_xor(m, off, 32));

    // B: exp + segment sum
    float ssum = 0.f;
    for (int idx = start + lane; idx < end; idx += 32) {
        float ex = __expf(tl[idx] - m);
        tl[idx] = ex;
        ssum += ex;
    }
    #pragma unroll
    for (int off = 16; off; off >>= 1) ssum += __shfl_xor(ssum, off, 32);
    const float inv = 1.f / (ssum + 1e-16f);

    // C: weighted accumulation in registers (o = lane, lane+32, lane+64, lane+96)
    float acc0 = 0.f, acc1 = 0.f, acc2 = 0.f, acc3 = 0.f;
    for (int j = start; j < end; ++j) {
        int e = eord[j];
        float alpha = tl[j] * inv;
        float4 a = ((const float4*)att)[et[e]];
        const float* xbs = xb + (size_t)src[e] * BD;     // L2-resident gather
        int o = lane;
        acc0 += alpha * (a.x * xbs[o] + a.y * xbs[DIMD + o]
                       + a.z * xbs[2*DIMD + o] + a.w * xbs[3*DIMD + o]);
        o = lane + 32;
        acc1 += alpha * (a.x * xbs[o] + a.y * xbs[DIMD + o]
                       + a.z * xbs[2*DIMD + o] + a.w * xbs[3*DIMD + o]);
        o = lane + 64;
        acc2 += alpha * (a.x * xbs[o] + a.y * xbs[DIMD + o]
                       + a.z * xbs[2*DIMD + o] + a.w * xbs[3*DIMD + o]);
        if (lane < 4) {
            o = lane + 96;
            acc3 += alpha * (a.x * xbs[o] + a.y * xbs[DIMD + o]
                           + a.z * xbs[2*DIMD + o] + a.w * xbs[3*DIMD + o]);
        }
    }

    float* od = out + (size_t)d * DIMD;
    float v;
    v = acc0 + bias[lane];      if (doRelu) v = fmaxf(v, 0.f); od[lane]      = v;
    v = acc1 + bias[lane + 32]; if (doRelu) v = fmaxf(v, 0.f); od[lane + 32] = v;
    v = acc2 + bias[lane + 64]; if (doRelu) v = fmaxf(v, 0.f); od[lane + 64] = v;
    if (lane < 4) {
        v = acc3 + bias[lane + 96];
        if (doRelu) v = fmaxf(v, 0.f);
        od[lane + 96] = v;
    }
}

// ---------------------------------------------------------------------------
extern "C" void kernel_launch(void* const* d_in, const int* in_sizes, int n_in,
                              void* d_out, int out_size, void* d_ws, size_t ws_size,
                              hipStream_t stream) {
    const int*   entity  = (const int*)  d_in[0];
    const int*   eidx    = (const int*)  d_in[1];   // [2,E] flat
    const int*   etype   = (const int*)  d_in[2];
    //           edge_norm (d_in[3]) is unused by the reference
    const float* emb     = (const float*)d_in[4];
    const float* basis1  = (const float*)d_in[5];
    const float* att1    = (const float*)d_in[6];
    const float* q1      = (const float*)d_in[7];
    const float* k1      = (const float*)d_in[8];
    const float* bias1   = (const float*)d_in[9];
    const float* basis2  = (const float*)d_in[10];
    const float* att2    = (const float*)d_in[11];
    const float* q2      = (const float*)d_in[12];
    const float* k2      = (const float*)d_in[13];
    const float* bias2   = (const float*)d_in[14];

    const int N = in_sizes[0];                 // 50000
    const int E = in_sizes[2];                 // 400000
    const int* src = eidx;
    const int* dst = eidx + E;

    // workspace carve-up
    float* xb   = (float*)d_ws;                     // N*400
    float* h1   = xb + (size_t)N * BD;              // N*100 (layer-1 out = layer-2 in)
    float* xq   = h1 + (size_t)N * DIMD;            // N*4
    float* xk   = xq + (size_t)N * 4;               // N*4
    float* tl   = xk + (size_t)N * 4;               // E
    float* qb   = tl + (size_t)E;                   // 400
    float* kb   = qb + BD;                          // 400
    int* cnt    = (int*)(kb + BD);                  // N
    int* rowptr = cnt + N;                          // N+1
    int* cursor = rowptr + N + 1;                   // N
    int* eord   = cursor + N;                       // E

    const int nRowTiles  = N / 16;                  // 3125 (divides evenly)
    const int gemmBlocks = (nRowTiles * 5 + 7) / 8; // 8 waves / 256-thread block
    const int eBlocks    = (E + 255) / 256;
    const int nBlocks    = (N + 255) / 256;
    const int aggBlocks  = (N + 7) / 8;             // wave per dst node

    // ---- CSR build (shared by both layers) ----
    rgat_csr_zero<<<nBlocks, 256, 0, stream>>>(cnt, N);
    rgat_csr_hist<<<eBlocks, 256, 0, stream>>>(dst, cnt, E);
    rgat_csr_scan<<<1, 1024, 0, stream>>>(cnt, rowptr, cursor, N);
    rgat_csr_fill<<<eBlocks, 256, 0, stream>>>(dst, cursor, eord, E);

    // ---- layer 1 (input = emb[entity], relu) ----
    rgat_qbkb     <<<2, 256, 0, stream>>>(basis1, q1, k1, qb, kb);
    rgat_gemm_xb  <<<gemmBlocks, 256, 0, stream>>>(emb, entity, basis1, xb, nRowTiles);
    rgat_xqxk     <<<nBlocks, 256, 0, stream>>>(emb, entity, qb, kb, xq, xk, N);
    rgat_aggregate<<<aggBlocks, 256, 0, stream>>>(rowptr, eord, src, etype, att1,
                                                  xq, xk, xb, bias1, tl, h1, N, 1);

    // ---- layer 2 (input = h1, no relu, output = d_out) ----
    float* out = (float*)d_out;
    rgat_qbkb     <<<2, 256, 0, stream>>>(basis2, q2, k2, qb, kb);
    rgat_gemm_xb  <<<gemmBlocks, 256, 0, stream>>>(h1, nullptr, basis2, xb, nRowTiles);
    rgat_xqxk     <<<nBlocks, 256, 0, stream>>>(h1, nullptr, qb, kb, xq, xk, N);
    rgat_aggregate<<<aggBlocks, 256, 0, stream>>>(rowptr, eord, src, etype, att2,
                                                  xq, xk, xb, bias2, tl, out, N, 0);
}